// GaussianSplatMomentMatching_2190433320967
// MI455X (gfx1250) — compile-verified
//
#include <hip/hip_runtime.h>
#include <stdint.h>
#include <math.h>

#define CDIM 59
#define FEAT 48
#define BLK  128
#define FULL_CHUNKS ((BLK * CDIM) / 4)          // 1888 16-B chunks per plane per full block
#define FULL_ITERS  (FULL_CHUNKS / BLK)         // 14
#define FULL_TAIL   (FULL_CHUNKS % BLK)         // 96

// ---- gfx1250 async global<->LDS DMA helpers (ASYNCcnt path) ----------------
// Streaming data is single-touch and larger than the 192MB L2 -> non-temporal.

static __device__ __forceinline__ uint32_t lds_addr32(const void* p) {
    // generic shared pointer: low 32 bits are the LDS byte offset
    return (uint32_t)(uintptr_t)p;
}

static __device__ __forceinline__ void async_load_b128(uint32_t lds, uint32_t off, const void* base) {
    asm volatile("global_load_async_to_lds_b128 %0, %1, %2 th:TH_LOAD_NT"
                 :: "v"(lds), "v"(off), "s"(base) : "memory");
}
static __device__ __forceinline__ void async_load_b32(uint32_t lds, uint32_t off, const void* base) {
    asm volatile("global_load_async_to_lds_b32 %0, %1, %2 th:TH_LOAD_NT"
                 :: "v"(lds), "v"(off), "s"(base) : "memory");
}
static __device__ __forceinline__ void async_store_b128(uint32_t lds, uint32_t off, void* base) {
    asm volatile("global_store_async_from_lds_b128 %0, %1, %2 th:TH_STORE_NT"
                 :: "v"(off), "v"(lds), "s"(base) : "memory");
}
static __device__ __forceinline__ void async_store_b32(uint32_t lds, uint32_t off, void* base) {
    asm volatile("global_store_async_from_lds_b32 %0, %1, %2 th:TH_STORE_NT"
                 :: "v"(off), "v"(lds), "s"(base) : "memory");
}
static __device__ __forceinline__ void wait_async0() {
    asm volatile("s_wait_asynccnt 0" ::: "memory");
}

// ---- per-point math --------------------------------------------------------

// Accumulate w * (R diag(s^2) R^T + d d^T) into symmetric S {xx,xy,xz,yy,yz,zz}
static __device__ __forceinline__ void accum_sigma(const float* r, float wk,
                                                   float mux, float muy, float muz,
                                                   float S[6]) {
    float qw = r[3], qx = r[4], qy = r[5], qz = r[6];
    float inv = 1.0f / sqrtf(qw*qw + qx*qx + qy*qy + qz*qz);
    qw *= inv; qx *= inv; qy *= inv; qz *= inv;
    float R00 = 1.f - 2.f*(qy*qy + qz*qz);
    float R01 = 2.f*(qx*qy - qw*qz);
    float R02 = 2.f*(qx*qz + qw*qy);
    float R10 = 2.f*(qx*qy + qw*qz);
    float R11 = 1.f - 2.f*(qx*qx + qz*qz);
    float R12 = 2.f*(qy*qz - qw*qx);
    float R20 = 2.f*(qx*qz - qw*qy);
    float R21 = 2.f*(qy*qz + qw*qx);
    float R22 = 1.f - 2.f*(qx*qx + qy*qy);
    float sx = r[7], sy = r[8], sz = r[9];
    float s2x = sx*sx, s2y = sy*sy, s2z = sz*sz;
    float dx = r[0] - mux, dy = r[1] - muy, dz = r[2] - muz;
    S[0] += wk*(s2x*R00*R00 + s2y*R01*R01 + s2z*R02*R02 + dx*dx);
    S[1] += wk*(s2x*R00*R10 + s2y*R01*R11 + s2z*R02*R12 + dx*dy);
    S[2] += wk*(s2x*R00*R20 + s2y*R01*R21 + s2z*R02*R22 + dx*dz);
    S[3] += wk*(s2x*R10*R10 + s2y*R11*R11 + s2z*R12*R12 + dy*dy);
    S[4] += wk*(s2x*R10*R20 + s2y*R11*R21 + s2z*R12*R22 + dy*dz);
    S[5] += wk*(s2x*R20*R20 + s2y*R21*R21 + s2z*R22*R22 + dz*dz);
}

// Eigenvector of symmetric S for eigenvalue lam, via largest cross product of rows of (S - lam I)
static __device__ __forceinline__ void eigvec3(const float S[6], float lam, float v[3]) {
    float r0x = S[0]-lam, r0y = S[1],     r0z = S[2];
    float r1x = S[1],     r1y = S[3]-lam, r1z = S[4];
    float r2x = S[2],     r2y = S[4],     r2z = S[5]-lam;
    float ax = r0y*r1z - r0z*r1y, ay = r0z*r1x - r0x*r1z, az = r0x*r1y - r0y*r1x;
    float bx = r0y*r2z - r0z*r2y, by = r0z*r2x - r0x*r2z, bz = r0x*r2y - r0y*r2x;
    float cx = r1y*r2z - r1z*r2y, cy = r1z*r2x - r1x*r2z, cz = r1x*r2y - r1y*r2x;
    float na = ax*ax + ay*ay + az*az;
    float nb = bx*bx + by*by + bz*bz;
    float nc = cx*cx + cy*cy + cz*cz;
    float mx = ax, my = ay, mz = az, mn = na;
    if (nb > mn) { mx = bx; my = by; mz = bz; mn = nb; }
    if (nc > mn) { mx = cx; my = cy; mz = cz; mn = nc; }
    if (mn < 1e-30f) { v[0] = 1.f; v[1] = 0.f; v[2] = 0.f; return; }
    float inv = rsqrtf(mn);
    v[0] = mx*inv; v[1] = my*inv; v[2] = mz*inv;
}

// ---- kernel ----------------------------------------------------------------

__global__ void __launch_bounds__(BLK)
gsplat_moment_kernel(const float* __restrict__ x, float* __restrict__ out,
                     int n, long long planeStride) {
    __shared__ float sm[2][BLK * CDIM];   // 60,416 B: K=0 and K=1 staging (out reuses plane 0)

    const int tid = threadIdx.x;
    const long long basePt = (long long)blockIdx.x * BLK;
    int cnt = (int)(n - basePt); if (cnt > BLK) cnt = BLK;
    const bool full = (cnt == BLK);
    const int nflt   = cnt * CDIM;
    const int nchunk = nflt >> 2;            // 16-B chunks
    const int rem    = nflt & 3;             // tail floats (0 for this problem's shapes)

    // Stage both K planes into LDS via async DMA (coalesced b128, non-temporal).
    for (int k = 0; k < 2; ++k) {
        const float* gsrc = x + (long long)k * planeStride + basePt * CDIM;
        uint32_t lbase = lds_addr32(&sm[k][0]);
        if (full) {
            // constant trip count: dense run of async issues, no per-iter mask churn
            #pragma unroll
            for (int i = 0; i < FULL_ITERS; ++i) {
                uint32_t c = (uint32_t)(tid + i * BLK) * 16u;
                async_load_b128(lbase + c, c, gsrc);
            }
            if (tid < FULL_TAIL) {
                uint32_t c = (uint32_t)(tid + FULL_ITERS * BLK) * 16u;
                async_load_b128(lbase + c, c, gsrc);
            }
        } else {
            for (int c = tid; c < nchunk; c += BLK)
                async_load_b128(lbase + (uint32_t)c * 16u, (uint32_t)c * 16u, gsrc);
            if (tid < rem)
                async_load_b32(lbase + (uint32_t)nchunk * 16u + (uint32_t)tid * 4u,
                               (uint32_t)nchunk * 16u + (uint32_t)tid * 4u, gsrc);
        }
    }
    wait_async0();
    __syncthreads();

    const float eps = 1.1920929e-07f;

    if (tid < cnt) {
        float* r0 = &sm[0][tid * CDIM];     // stride 59 (odd) -> bank-conflict free
        float* r1 = &sm[1][tid * CDIM];

        // weights
        float o0 = r0[10], o1 = r1[10];
        float osum = o0 + o1;
        float wsum = fmaxf(osum, eps);
        float w0 = o0 / wsum, w1 = o1 / wsum;

        // mean
        float mux = w0*r0[0] + w1*r1[0];
        float muy = w0*r0[1] + w1*r1[1];
        float muz = w0*r0[2] + w1*r1[2];

        // moment-matched covariance (symmetric)
        float S[6] = {0.f, 0.f, 0.f, 0.f, 0.f, 0.f};
        accum_sigma(r0, w0, mux, muy, muz, S);
        accum_sigma(r1, w1, mux, muy, muz, S);

        // analytic symmetric 3x3 eigendecomposition (ascending)
        float a00 = S[0], a01 = S[1], a02 = S[2], a11 = S[3], a12 = S[4], a22 = S[5];
        float p1 = a01*a01 + a02*a02 + a12*a12;
        float qm = (a00 + a11 + a22) * (1.f/3.f);
        float b00 = a00 - qm, b11 = a11 - qm, b22 = a22 - qm;
        float p2 = b00*b00 + b11*b11 + b22*b22 + 2.f*p1;

        float e0, e1, e2;
        float V0[3], V1[3], V2[3];
        if (p2 <= 1e-26f) {
            // (near-)isotropic: Sigma ~ qm * I
            e0 = e1 = e2 = qm;
            V0[0]=1.f; V0[1]=0.f; V0[2]=0.f;
            V1[0]=0.f; V1[1]=1.f; V1[2]=0.f;
            V2[0]=0.f; V2[1]=0.f; V2[2]=1.f;
        } else {
            float p    = sqrtf(p2 * (1.f/6.f));
            float invp = 1.f / p;
            float c00 = b00*invp, c01 = a01*invp, c02 = a02*invp;
            float c11 = b11*invp, c12 = a12*invp, c22 = b22*invp;
            float detB = c00*(c11*c22 - c12*c12)
                       - c01*(c01*c22 - c12*c02)
                       + c02*(c01*c12 - c11*c02);
            float rr = fminf(fmaxf(0.5f*detB, -1.f), 1.f);
            float phi = acosf(rr) * (1.f/3.f);
            e2 = qm + 2.f*p*cosf(phi);
            e0 = qm + 2.f*p*cosf(phi + 2.0943951023931953f);
            e1 = 3.f*qm - e0 - e2;
            eigvec3(S, e0, V0);
            eigvec3(S, e2, V2);
            // middle eigenvector: orthogonal completion (det = +1 by construction)
            V1[0] = V2[1]*V0[2] - V2[2]*V0[1];
            V1[1] = V2[2]*V0[0] - V2[0]*V0[2];
            V1[2] = V2[0]*V0[1] - V2[1]*V0[0];
            float n1 = V1[0]*V1[0] + V1[1]*V1[1] + V1[2]*V1[2];
            float in1 = (n1 > 1e-30f) ? rsqrtf(n1) : 0.f;
            V1[0] *= in1; V1[1] *= in1; V1[2] *= in1;
        }

        // det-sign flip of last column (mirrors reference; no-op when det=+1)
        float det = V0[0]*(V1[1]*V2[2] - V1[2]*V2[1])
                  - V0[1]*(V1[0]*V2[2] - V1[2]*V2[0])
                  + V0[2]*(V1[0]*V2[1] - V1[1]*V2[0]);
        if (det < 0.f) { V2[0] = -V2[0]; V2[1] = -V2[1]; V2[2] = -V2[2]; }

        // rotation matrix with eigenvector columns
        float R00 = V0[0], R01 = V1[0], R02 = V2[0];
        float R10 = V0[1], R11 = V1[1], R12 = V2[1];
        float R20 = V0[2], R21 = V1[2], R22 = V2[2];

        // rotmat -> quaternion (4-candidate argmax, as in reference)
        float t0 = 1.f + R00 + R11 + R22;
        float t1 = 1.f + R00 - R11 - R22;
        float t2 = 1.f - R00 + R11 - R22;
        float t3 = 1.f - R00 - R11 + R22;
        float qw, qx, qy, qz;
        if (t0 >= t1 && t0 >= t2 && t0 >= t3) {
            float s = 2.f*sqrtf(fmaxf(t0, eps)), is = 1.f/s;
            qw = 0.25f*s; qx = (R21-R12)*is; qy = (R02-R20)*is; qz = (R10-R01)*is;
        } else if (t1 >= t2 && t1 >= t3) {
            float s = 2.f*sqrtf(fmaxf(t1, eps)), is = 1.f/s;
            qw = (R21-R12)*is; qx = 0.25f*s; qy = (R01+R10)*is; qz = (R02+R20)*is;
        } else if (t2 >= t3) {
            float s = 2.f*sqrtf(fmaxf(t2, eps)), is = 1.f/s;
            qw = (R02-R20)*is; qx = (R01+R10)*is; qy = 0.25f*s; qz = (R12+R21)*is;
        } else {
            float s = 2.f*sqrtf(fmaxf(t3, eps)), is = 1.f/s;
            qw = (R10-R01)*is; qx = (R02+R20)*is; qy = (R12+R21)*is; qz = 0.25f*s;
        }
        float qn = 1.f / sqrtf(qw*qw + qx*qx + qy*qy + qz*qz);
        qw *= qn; qx *= qn; qy *= qn; qz *= qn;
        if (qw < 0.f) { qw = -qw; qx = -qx; qy = -qy; qz = -qz; }

        // write output row into plane-0 LDS buffer (all overwritten slots already consumed)
        r0[0] = mux; r0[1] = muy; r0[2] = muz;
        r0[3] = qw;  r0[4] = qx;  r0[5] = qy;  r0[6] = qz;
        r0[7] = sqrtf(fmaxf(e0, eps));
        r0[8] = sqrtf(fmaxf(e1, eps));
        r0[9] = sqrtf(fmaxf(e2, eps));
        r0[10] = fminf(fmaxf(osum, eps), 1.0f);
        #pragma unroll 4
        for (int j = 0; j < FEAT; ++j)
            r0[11 + j] = w0*r0[11 + j] + w1*r1[11 + j];
    }
    __syncthreads();

    // Drain output block LDS -> HBM via async DMA (coalesced b128, non-temporal).
    {
        float* gdst = out + basePt * CDIM;
        uint32_t lbase = lds_addr32(&sm[0][0]);
        if (full) {
            #pragma unroll
            for (int i = 0; i < FULL_ITERS; ++i) {
                uint32_t c = (uint32_t)(tid + i * BLK) * 16u;
                async_store_b128(lbase + c, c, gdst);
            }
            if (tid < FULL_TAIL) {
                uint32_t c = (uint32_t)(tid + FULL_ITERS * BLK) * 16u;
                async_store_b128(lbase + c, c, gdst);
            }
        } else {
            for (int c = tid; c < nchunk; c += BLK)
                async_store_b128(lbase + (uint32_t)c * 16u, (uint32_t)c * 16u, gdst);
            if (tid < rem)
                async_store_b32(lbase + (uint32_t)nchunk * 16u + (uint32_t)tid * 4u,
                                (uint32_t)nchunk * 16u + (uint32_t)tid * 4u, gdst);
        }
        wait_async0();
    }
}

// ---- host launch -----------------------------------------------------------

extern "C" void kernel_launch(void* const* d_in, const int* in_sizes, int n_in,
                              void* d_out, int out_size, void* d_ws, size_t ws_size,
                              hipStream_t stream) {
    (void)n_in; (void)d_ws; (void)ws_size; (void)out_size;
    const float* x = (const float*)d_in[0];
    float* out = (float*)d_out;
    const int n = in_sizes[0] / (2 * CDIM);             // 500000
    const long long planeStride = (long long)n * CDIM;  // floats per K plane
    dim3 grid((unsigned)((n + BLK - 1) / BLK)), block(BLK);
    gsplat_moment_kernel<<<grid, block, 0, stream>>>(x, out, n, planeStride);
}